// CTProjector3DModule_32306744000775
// MI455X (gfx1250) — compile-verified
//
#include <hip/hip_runtime.h>
#include <hip/hip_bf16.h>
#include <stdint.h>

#define N_VOXD 128
#define SLAB_ELEMS (N_VOXD * N_VOXD)   /* 16384 floats */
#define SLAB_BYTES (SLAB_ELEMS * 4)    /* 65536 bytes  */
#define BIGF 1e9f
#define EPSF 1e-12f

#if defined(__gfx1250__) && __has_builtin(__builtin_amdgcn_tensor_load_to_lds) && \
    __has_builtin(__builtin_amdgcn_s_wait_tensorcnt)
#define USE_TDM 1
#else
#define USE_TDM 0
#endif

#if USE_TDM
typedef unsigned int u32x4 __attribute__((ext_vector_type(4)));
typedef int          i32x4 __attribute__((ext_vector_type(4)));
typedef int          i32x8 __attribute__((ext_vector_type(8)));

// Build a D# describing a 2-D 128x128 tile of 4-byte elements (one x-slab of
// the volume) and issue TENSOR_LOAD_TO_LDS.  Tracked with TENSORcnt.
__device__ __forceinline__ void tdm_load_slab(uint32_t lds_byte_addr, const float* gsrc) {
  uint64_t ga = (uint64_t)(uintptr_t)gsrc;
  u32x4 g0;
  g0.x = 1u;                                         // count=1, user mode
  g0.y = lds_byte_addr;                              // lds_addr (bytes)
  g0.z = (uint32_t)(ga & 0xFFFFFFFFu);               // global_addr[31:0]
  g0.w = (uint32_t)((ga >> 32) & 0x01FFFFFFu)        // global_addr[56:32]
       | (2u << 30);                                 // type = 2 ("image")
  i32x8 g1;
  g1[0] = (int)(2u << 16);                           // data_size=2 (4B); wg_mask=0
  g1[1] = (int)((uint32_t)N_VOXD << 16);             // tensor_dim0[15:0] = 128
  g1[2] = (int)((uint32_t)N_VOXD << 16);             // tensor_dim0[31:16]=0 | tensor_dim1[15:0]=128
  g1[3] = (int)((uint32_t)N_VOXD << 16);             // tensor_dim1[31:16]=0 | tile_dim0=128
  g1[4] = (int)((uint32_t)N_VOXD);                   // tile_dim1=128, tile_dim2=0
  g1[5] = N_VOXD;                                    // tensor_dim0_stride = 128
  g1[6] = 0;                                         // stride0[47:32]=0 | stride1[15:0]=0
  g1[7] = 0;
  i32x4 z4 = {0, 0, 0, 0};
#if __has_include(<hip/amd_detail/amd_gfx1250_TDM.h>)
  i32x8 z8 = {0, 0, 0, 0, 0, 0, 0, 0};
  __builtin_amdgcn_tensor_load_to_lds(g0, g1, z4, z4, z8, 0);
#else
  __builtin_amdgcn_tensor_load_to_lds(g0, g1, z4, z4, 0);
#endif
}
#endif // USE_TDM

__global__ void __launch_bounds__(256)
ct_siddon_slab_kernel(const float* __restrict__ vol,
                      const float* __restrict__ Mmat,
                      const float* __restrict__ bvec,
                      const float* __restrict__ src,
                      const float* __restrict__ dst,
                      float* __restrict__ out,
                      int n_ray)
{
  extern __shared__ float smem[];            // 2 * 16384 floats (double buffer)
  const int tid = threadIdx.x;
  const int ray = blockIdx.x * 256 + tid;
  const bool active = ray < n_ray;
  const int r = active ? ray : 0;

  // ---- per-ray setup: Minv via adjugate, endpoints in voxel-index coords ----
  const float m00 = Mmat[0], m01 = Mmat[1], m02 = Mmat[2];
  const float m10 = Mmat[3], m11 = Mmat[4], m12 = Mmat[5];
  const float m20 = Mmat[6], m21 = Mmat[7], m22 = Mmat[8];
  const float det = m00 * (m11 * m22 - m12 * m21)
                  - m01 * (m10 * m22 - m12 * m20)
                  + m02 * (m10 * m21 - m11 * m20);
  const float idt = 1.0f / det;
  const float i00 = (m11 * m22 - m12 * m21) * idt, i01 = (m02 * m21 - m01 * m22) * idt,
              i02 = (m01 * m12 - m02 * m11) * idt;
  const float i10 = (m12 * m20 - m10 * m22) * idt, i11 = (m00 * m22 - m02 * m20) * idt,
              i12 = (m02 * m10 - m00 * m12) * idt;
  const float i20 = (m10 * m21 - m11 * m20) * idt, i21 = (m01 * m20 - m00 * m21) * idt,
              i22 = (m00 * m11 - m01 * m10) * idt;
  const float bx = bvec[0], by = bvec[1], bz = bvec[2];

  const float s0 = src[3 * r] - bx, s1 = src[3 * r + 1] - by, s2 = src[3 * r + 2] - bz;
  const float e0 = dst[3 * r] - bx, e1 = dst[3 * r + 1] - by, e2 = dst[3 * r + 2] - bz;
  const float p0x = i00 * s0 + i01 * s1 + i02 * s2;
  const float p0y = i10 * s0 + i11 * s1 + i12 * s2;
  const float p0z = i20 * s0 + i21 * s1 + i22 * s2;
  const float p1x = i00 * e0 + i01 * e1 + i02 * e2;
  const float p1y = i10 * e0 + i11 * e1 + i12 * e2;
  const float p1z = i20 * e0 + i21 * e1 + i22 * e2;
  const float dx = p1x - p0x, dy = p1y - p0y, dz = p1z - p0z;

  const float wx = dst[3 * r] - src[3 * r];
  const float wy = dst[3 * r + 1] - src[3 * r + 1];
  const float wz = dst[3 * r + 2] - src[3 * r + 2];
  const float ray_len = sqrtf(wx * wx + wy * wy + wz * wz);

  const bool parx = fabsf(dx) < EPSF, pary = fabsf(dy) < EPSF, parz = fabsf(dz) < EPSF;
  const float invdx = 1.0f / (parx ? 1.0f : dx);
  const float invdy = 1.0f / (pary ? 1.0f : dy);
  const float invdz = 1.0f / (parz ? 1.0f : dz);

  // global ray/box interval [tmin, tmax] (clamped to [0,1] like the reference)
  float tmin = 0.0f, tmax = 1.0f;
  {
    float t0 = (0.0f - p0x) * invdx, t1 = (128.0f - p0x) * invdx;
    float lo = fminf(t0, t1), hi = fmaxf(t0, t1);
    if (parx) { bool in = (p0x >= 0.0f) && (p0x <= 128.0f); lo = in ? 0.0f : BIGF; hi = in ? 1.0f : -BIGF; }
    tmin = fmaxf(tmin, lo); tmax = fminf(tmax, hi);
  }
  {
    float t0 = (0.0f - p0y) * invdy, t1 = (128.0f - p0y) * invdy;
    float lo = fminf(t0, t1), hi = fmaxf(t0, t1);
    if (pary) { bool in = (p0y >= 0.0f) && (p0y <= 128.0f); lo = in ? 0.0f : BIGF; hi = in ? 1.0f : -BIGF; }
    tmin = fmaxf(tmin, lo); tmax = fminf(tmax, hi);
  }
  {
    float t0 = (0.0f - p0z) * invdz, t1 = (128.0f - p0z) * invdz;
    float lo = fminf(t0, t1), hi = fmaxf(t0, t1);
    if (parz) { bool in = (p0z >= 0.0f) && (p0z <= 128.0f); lo = in ? 0.0f : BIGF; hi = in ? 1.0f : -BIGF; }
    tmin = fmaxf(tmin, lo); tmax = fminf(tmax, hi);
  }
  if (!((tmax > tmin) && active)) { tmin = BIGF; tmax = -BIGF; }  // empty interval; still run barriers

  float sum = 0.0f;

  // ---- prologue: stage slabs 0 and 1 ----
#if USE_TDM
  if (tid < 32) {                       // wave 0 drives the TDM; EXEC ignored by tensor ops
    tdm_load_slab(0u, vol);
    tdm_load_slab((uint32_t)SLAB_BYTES, vol + SLAB_ELEMS);
  }
#else
  for (int i = tid; i < SLAB_ELEMS; i += 256) smem[i] = vol[i];
  for (int i = tid; i < SLAB_ELEMS; i += 256) smem[SLAB_ELEMS + i] = vol[SLAB_ELEMS + i];
#endif

  // ---- march over x-slabs; rays here are x-major so <=3 segments per slab ----
#pragma unroll 1
  for (int xi = 0; xi < N_VOXD; ++xi) {
#if USE_TDM
    if (tid < 32) {
      if (xi < N_VOXD - 1) __builtin_amdgcn_s_wait_tensorcnt(1);  // slab xi done, xi+1 in flight
      else                 __builtin_amdgcn_s_wait_tensorcnt(0);
    }
#endif
    __syncthreads();                    // slab xi visible to the whole block

    const float* sl = &smem[(xi & 1) * SLAB_ELEMS];
    const float xf = (float)xi;
    float ta, tb;
    if (!parx) {
      float t0 = (xf - p0x) * invdx, t1 = (xf + 1.0f - p0x) * invdx;
      ta = fminf(t0, t1); tb = fmaxf(t0, t1);
    } else {
      bool in = (p0x >= xf) && (p0x <= xf + 1.0f);
      ta = in ? -BIGF : BIGF; tb = in ? BIGF : -BIGF;
    }
    ta = fmaxf(ta, tmin); tb = fminf(tb, tmax);

    if (tb > ta + 1e-9f) {
      float tcur = ta;
#pragma unroll 1
      for (int k = 0; k < 8; ++k) {     // x-major rays: at most ~3 iterations
        const float tref = tcur + 1e-6f;
        const float yc = p0y + tref * dy;
        const float zc = p0z + tref * dz;
        const float tny = pary ? BIGF
            : (((dy > 0.0f) ? (floorf(yc) + 1.0f) : (ceilf(yc) - 1.0f)) - p0y) * invdy;
        const float tnz = parz ? BIGF
            : (((dz > 0.0f) ? (floorf(zc) + 1.0f) : (ceilf(zc) - 1.0f)) - p0z) * invdz;
        float tend = fminf(fminf(tny, tnz), tb);
        tend = fmaxf(tend, tcur);
        const float tm = 0.5f * (tcur + tend);
        const int iy = (int)floorf(p0y + tm * dy);
        const int iz = (int)floorf(p0z + tm * dz);
        const float dt = tend - tcur;
        if (iy >= 0 && iy < N_VOXD && iz >= 0 && iz < N_VOXD && dt > 0.0f)
          sum += dt * sl[iy * N_VOXD + iz];   // LDS gather
        tcur = tend;
        if (tcur >= tb - 1e-7f) break;
      }
    }

    __syncthreads();                    // everyone done reading buf[xi&1]
#if USE_TDM
    if (tid < 32 && xi + 2 < N_VOXD)
      tdm_load_slab((uint32_t)((xi & 1) * SLAB_BYTES),
                    vol + (size_t)(xi + 2) * SLAB_ELEMS);
#else
    if (xi + 2 < N_VOXD) {
      const float* g = vol + (size_t)(xi + 2) * SLAB_ELEMS;
      float* dl = &smem[(xi & 1) * SLAB_ELEMS];
      for (int i = tid; i < SLAB_ELEMS; i += 256) dl[i] = g[i];
    }
#endif
  }

  if (active) out[ray] = sum * ray_len;
}

extern "C" void kernel_launch(void* const* d_in, const int* in_sizes, int n_in,
                              void* d_out, int out_size, void* d_ws, size_t ws_size,
                              hipStream_t stream) {
  const float* vol = (const float*)d_in[0];
  const float* M   = (const float*)d_in[1];
  const float* b   = (const float*)d_in[2];
  const float* src = (const float*)d_in[3];
  const float* dst = (const float*)d_in[4];
  float* out = (float*)d_out;

  const int n_ray = in_sizes[3] / 3;
  const int block = 256;
  const int grid = (n_ray + block - 1) / block;
  const size_t shmem = 2 * (size_t)SLAB_BYTES;   // 128 KB double buffer in 320 KB LDS

  hipLaunchKernelGGL(ct_siddon_slab_kernel, dim3(grid), dim3(block), shmem, stream,
                     vol, M, b, src, dst, out, n_ray);
}